// Luong_attention_76484777607207
// MI455X (gfx1250) — compile-verified
//
#include <hip/hip_runtime.h>
#include <hip/hip_bf16.h>
#include <math.h>

// ---------------------------------------------------------------------------
// Luong attention for MI455X (gfx1250, wave32, WMMA + async-to-LDS).
//   context, attn = softmax(mask((q @ W) @ v^T)) ; context = attn @ v
// Memory-bound on the 134MB attn matrix -> f32 end-to-end, f32 WMMA.
// Staging uses GLOBAL_LOAD_ASYNC_TO_LDS_B128 (ASYNCcnt) with LDS double
// buffering so the next K-chunk streams in while WMMAs run on the current one.
// ---------------------------------------------------------------------------

typedef __attribute__((ext_vector_type(2))) float v2f;
typedef __attribute__((ext_vector_type(8))) float v8f;

#define BM 64          // block tile rows (M)
#define BN 128         // block tile cols (N)
#define BK 32          // K chunk staged through LDS
#define AS_STRIDE 36   // BK+4: rows 16B-aligned for b128 LDS writes, bank-clean
#define AS_FLOATS (BM * AS_STRIDE)          // 2304 floats per buffer
#define BS_STRIDE_T 36                      // BT:  Bs[n][k], 128 rows
#define BS_STRIDE_N 132                     // !BT: Bs[k][n], 32 rows (BN+4)
#define BS_FLOATS 4608                      // max(128*36, 32*132)
#define SMEM_FLOATS (2 * AS_FLOATS + 2 * BS_FLOATS)   // 13824 floats = 54 KB

// 256 threads = 8 wave32s in a 2(M) x 4(N) grid; each wave computes a 32x32
// sub-tile as 2x2 WMMA 16x16 tiles -> 4 wmma share 4 fragment loads per K-step.
//
// WMMA f32 16x16x4 fragments (ISA 7.12.2):
//   A 16x4:  lanes 0-15: M=lane, v[0]=K, v[1]=K+1 ; lanes 16-31: K+2/K+3
//   B 4x16:  lanes 0-15: N=lane, v[0]=K, v[1]=K+1 ; lanes 16-31: K+2/K+3
//   C/D:     VGPR r: lanes 0-15 -> M=r ; lanes 16-31 -> M=r+8
template <bool BT>
__global__ __launch_bounds__(256) void gemm_wmma_f32(
    const float* __restrict__ A, const float* __restrict__ B,
    float* __restrict__ C, int M, int N, int K,
    long long sA, long long sB, long long sC)
{
    __shared__ float smem[SMEM_FLOATS];
    // Flat aperture addr truncated to 32 bits == LDS byte offset (ISA 10.2).
    const unsigned ldsBase = (unsigned)(size_t)(&smem[0]);

    const float* __restrict__ Ab = A + (size_t)blockIdx.z * sA;
    const float* __restrict__ Bb = B + (size_t)blockIdx.z * sB;
    float*       __restrict__ Cb = C + (size_t)blockIdx.z * sC;

    const int m0   = blockIdx.y * BM;
    const int n0   = blockIdx.x * BN;
    const int tid  = threadIdx.x;
    const int lane = tid & 31;
    const int wave = tid >> 5;
    const int wmBase = (wave >> 2) << 5;   // 0 | 32
    const int wnBase = (wave & 3) << 5;    // 0 | 32 | 64 | 96
    const int mr   = lane & 15;
    const int kh   = (lane >> 4) << 1;     // upper half-wave holds K+2/K+3

    // Issue one 16B async copy: memory -> LDS, no VGPR round-trip (ASYNCcnt).
    auto asyncCopy = [&](unsigned ldsOff, const float* src) {
        asm volatile("global_load_async_to_lds_b128 %0, %1, off"
                     :: "v"(ldsOff), "v"(src) : "memory");
    };

    // Stage one BK-chunk of A (64x32) and B (128x32 | 32x128): 6 b128/thread.
    auto issueTiles = [&](int k0, int buf) {
        #pragma unroll
        for (int it = 0; it < 2; ++it) {                 // A: 512 float4
            int f = tid + (it << 8);
            int row = f >> 3, c4 = (f & 7) << 2;
            asyncCopy(ldsBase + ((unsigned)(buf * AS_FLOATS + row * AS_STRIDE + c4) << 2),
                      Ab + (size_t)(m0 + row) * K + k0 + c4);
        }
        const unsigned bsBase = 2 * AS_FLOATS + buf * BS_FLOATS;
        #pragma unroll
        for (int it = 0; it < 4; ++it) {                 // B: 1024 float4
            int f = tid + (it << 8);
            if (BT) {       // B memory is NxK (values): tile rows contiguous
                int n = f >> 3, c4 = (f & 7) << 2;
                asyncCopy(ldsBase + ((bsBase + n * BS_STRIDE_T + c4) << 2),
                          Bb + (size_t)(n0 + n) * K + k0 + c4);
            } else {        // B memory is KxN row-major
                int k = f >> 5, c4 = (f & 31) << 2;
                asyncCopy(ldsBase + ((bsBase + k * BS_STRIDE_N + c4) << 2),
                          Bb + (size_t)(k0 + k) * N + n0 + c4);
            }
        }
    };

    v8f acc00 = {0,0,0,0,0,0,0,0}, acc01 = {0,0,0,0,0,0,0,0};
    v8f acc10 = {0,0,0,0,0,0,0,0}, acc11 = {0,0,0,0,0,0,0,0};

    issueTiles(0, 0);
    int buf = 0;
    for (int k0 = 0; k0 < K; k0 += BK) {
        if (k0 + BK < K) {
            issueTiles(k0 + BK, buf ^ 1);
            // Async loads retire in order: <=6 outstanding means the previous
            // chunk's 6 copies are done; the 6 just issued keep streaming.
            asm volatile("s_wait_asynccnt 6" ::: "memory");
        } else {
            asm volatile("s_wait_asynccnt 0" ::: "memory");
        }
        __syncthreads();

        const int aBase = buf * AS_FLOATS;
        const int bBase = 2 * AS_FLOATS + buf * BS_FLOATS;
        const int rA0 = aBase + (wmBase + mr) * AS_STRIDE;
        const int rA1 = aBase + (wmBase + 16 + mr) * AS_STRIDE;

        #pragma unroll
        for (int kk = 0; kk < BK; kk += 4) {
            const int ka = kk + kh;
            v2f a0, a1, b0, b1;
            a0.x = smem[rA0 + ka]; a0.y = smem[rA0 + ka + 1];
            a1.x = smem[rA1 + ka]; a1.y = smem[rA1 + ka + 1];
            if (BT) {
                const int rB0 = bBase + (wnBase + mr) * BS_STRIDE_T;
                const int rB1 = bBase + (wnBase + 16 + mr) * BS_STRIDE_T;
                b0.x = smem[rB0 + ka]; b0.y = smem[rB0 + ka + 1];
                b1.x = smem[rB1 + ka]; b1.y = smem[rB1 + ka + 1];
            } else {
                const int rK0 = bBase + ka * BS_STRIDE_N;
                const int rK1 = rK0 + BS_STRIDE_N;
                b0.x = smem[rK0 + wnBase + mr];      b0.y = smem[rK1 + wnBase + mr];
                b1.x = smem[rK0 + wnBase + 16 + mr]; b1.y = smem[rK1 + wnBase + 16 + mr];
            }
            // 4 wmma per 4 fragment loads; 4 independent acc chains for ILP.
            acc00 = __builtin_amdgcn_wmma_f32_16x16x4_f32(false, a0, false, b0, (short)0, acc00, false, false);
            acc01 = __builtin_amdgcn_wmma_f32_16x16x4_f32(false, a0, false, b1, (short)0, acc01, false, false);
            acc10 = __builtin_amdgcn_wmma_f32_16x16x4_f32(false, a1, false, b0, (short)0, acc10, false, false);
            acc11 = __builtin_amdgcn_wmma_f32_16x16x4_f32(false, a1, false, b1, (short)0, acc11, false, false);
        }
        __syncthreads();
        buf ^= 1;
    }

    // Store: lanes 0-15 cover 16 consecutive cols of one row -> 64B coalesced.
    const int rsel = (lane >> 4) << 3;
    {
        const int rb = m0 + wmBase + rsel, cc = n0 + wnBase + mr;
        #pragma unroll
        for (int r = 0; r < 8; ++r) Cb[(size_t)(rb + r) * N + cc] = acc00[r];
    }
    {
        const int rb = m0 + wmBase + rsel, cc = n0 + wnBase + 16 + mr;
        #pragma unroll
        for (int r = 0; r < 8; ++r) Cb[(size_t)(rb + r) * N + cc] = acc01[r];
    }
    {
        const int rb = m0 + wmBase + 16 + rsel, cc = n0 + wnBase + mr;
        #pragma unroll
        for (int r = 0; r < 8; ++r) Cb[(size_t)(rb + r) * N + cc] = acc10[r];
    }
    {
        const int rb = m0 + wmBase + 16 + rsel, cc = n0 + wnBase + 16 + mr;
        #pragma unroll
        for (int r = 0; r < 8; ++r) Cb[(size_t)(rb + r) * N + cc] = acc11[r];
    }
}

// One block per attention row; S == 2048 -> 8 values cached per thread.
__global__ __launch_bounds__(256) void softmax_mask_rows(
    float* __restrict__ attn, const int* __restrict__ mask, int Q, int S)
{
    const int row   = blockIdx.x;          // 0 .. B*Q-1
    const int batch = row / Q;
    float* __restrict__ p = attn + (size_t)row * S;
    const int* __restrict__ mrow = mask + (size_t)batch * S;

    const int tid = threadIdx.x;
    float v[8];
    float lmax = -INFINITY;
    #pragma unroll
    for (int i = 0; i < 8; ++i) {
        int s   = tid + (i << 8);
        float x = p[s];
        x = (mrow[s] != 0) ? x : -INFINITY;
        v[i] = x;
        lmax = fmaxf(lmax, x);
    }
    __shared__ float red[256];
    red[tid] = lmax;
    __syncthreads();
    for (int off = 128; off > 0; off >>= 1) {
        if (tid < off) red[tid] = fmaxf(red[tid], red[tid + off]);
        __syncthreads();
    }
    const float rmax = red[0];
    __syncthreads();

    float lsum = 0.f;
    #pragma unroll
    for (int i = 0; i < 8; ++i) {
        float e = __expf(v[i] - rmax);
        v[i] = e;
        lsum += e;
    }
    red[tid] = lsum;
    __syncthreads();
    for (int off = 128; off > 0; off >>= 1) {
        if (tid < off) red[tid] += red[tid + off];
        __syncthreads();
    }
    const float inv = 1.0f / red[0];
    #pragma unroll
    for (int i = 0; i < 8; ++i)
        p[tid + (i << 8)] = v[i] * inv;
}

extern "C" void kernel_launch(void* const* d_in, const int* in_sizes, int n_in,
                              void* d_out, int out_size, void* d_ws, size_t ws_size,
                              hipStream_t stream)
{
    (void)in_sizes; (void)n_in; (void)out_size; (void)d_ws; (void)ws_size;

    const float* query  = (const float*)d_in[0];  // [B,Q,E]
    const float* values = (const float*)d_in[1];  // [B,S,E]
    const int*   amask  = (const int*)  d_in[2];  // [B,S]
    const float* W      = (const float*)d_in[3];  // [E,E]

    const int B = 8, Q = 2048, S = 2048, E = 512;

    float* context = (float*)d_out;                         // [B,Q,E]
    float* attn    = (float*)d_out + (size_t)B * Q * E;     // [B,Q,S]
    // tq has exactly context's shape and context isn't written until stage 4
    // (which no longer reads tq) -> reuse the context region as tq scratch.
    float* tq      = context;

    // 1) tq = query @ W  : single GEMM, M = B*Q (W shared across batch)
    gemm_wmma_f32<false>
        <<<dim3(E / BN, (B * Q) / BM, 1), 256, 0, stream>>>(
            query, W, tq, B * Q, E, E, 0, 0, 0);

    // 2) scores = tq @ values^T  per batch -> written into attn region
    gemm_wmma_f32<true>
        <<<dim3(S / BN, Q / BM, B), 256, 0, stream>>>(
            tq, values, attn, Q, S, E,
            (long long)Q * E, (long long)S * E, (long long)Q * S);

    // 3) masked softmax along S, in place
    softmax_mask_rows<<<B * Q, 256, 0, stream>>>(attn, amask, Q, S);

    // 4) context = attn @ values  per batch (overwrites the tq scratch)
    gemm_wmma_f32<false>
        <<<dim3(E / BN, Q / BM, B), 256, 0, stream>>>(
            attn, values, context, Q, E, S,
            (long long)Q * S, (long long)S * E, (long long)Q * E);
}